// CoreSnapshotEncoder_22849226015130
// MI455X (gfx1250) — compile-verified
//
#include <hip/hip_runtime.h>
#include <hip/hip_bf16.h>
#include <math.h>

// Problem constants (match the reference).
#define NQ     512          // NUM_QUBITS
#define NC     16           // NUM_CORES
#define CSZ    64           // CORE_SIZE
#define HID    128          // HIDDEN

typedef float v2f __attribute__((ext_vector_type(2)));
typedef float v8f __attribute__((ext_vector_type(8)));

// ---------------------------------------------------------------------------
// Kernel 1: a_norm[i][j] = A[i][j] / sqrt(deg_i * deg_j), deg = column sums.
// Tiny: one 256-thread block, one element per thread.
// ---------------------------------------------------------------------------
__global__ void anorm_kernel(const float* __restrict__ adj,
                             float* __restrict__ anorm) {
    int t = threadIdx.x;            // 0..255
    int i = t >> 4, j = t & 15;
    float degi = 0.f, degj = 0.f;
    #pragma unroll
    for (int k = 0; k < NC; ++k) {  // adjacency.sum(axis=0) -> column sums
        degi += adj[k * NC + i];
        degj += adj[k * NC + j];
    }
    float di = degi > 0.f ? 1.0f / sqrtf(degi) : 0.f;
    float dj = degj > 0.f ? 1.0f / sqrtf(degj) : 0.f;
    anorm[t] = adj[i * NC + j] * di * dj;
}

// ---------------------------------------------------------------------------
// Kernel 2: exact stable-order bucketing. One wave (32 threads) per batch.
// Lanes 0..15 each own a core and scan the 512 assignments in order
// (LDS broadcast read per iteration). Per-lane scalar counter -> no dynamic
// register indexing. First 64 hits per core are recorded; count is exact.
// ---------------------------------------------------------------------------
__global__ void __launch_bounds__(32)
scatter_kernel(const int* __restrict__ assign,   // (B, 512)
               int* __restrict__ cnts,           // (B, 16)
               int* __restrict__ lists) {        // (B, 16, 64)
    __shared__ int a_sh[NQ];
    const int b    = blockIdx.x;
    const int lane = threadIdx.x;

    const int* row = assign + (size_t)b * NQ;
    #pragma unroll
    for (int i = lane; i < NQ; i += 32) a_sh[i] = row[i];
    __syncthreads();

    if (lane < NC) {
        int cnt = 0;
        int* mylist = lists + ((size_t)b * NC + lane) * CSZ;
        for (int q = 0; q < NQ; ++q) {
            if (a_sh[q] == lane) {            // core 16 never occurs, excluded anyway
                if (cnt < CSZ) mylist[cnt] = q;
                ++cnt;
            }
        }
        cnts[b * NC + lane] = cnt;
    }
}

// ---------------------------------------------------------------------------
// Kernel 3: per batch (one 256-thread block):
//   Phase B: E[c][h] = max over listed qubits of emb[q][h]; max with 0 if cnt<64
//   Phase G: G = a_norm @ E              (fold the 16x16 mixing before the GEMM)
//   Phase C: out_b = G @ W^T + bias  via v_wmma_f32_16x16x4_f32 (fp32 WMMA)
// ---------------------------------------------------------------------------
__global__ void __launch_bounds__(256)
fused_kernel(const float* __restrict__ emb,    // (513, 128)
             const float* __restrict__ W,      // (128, 128) row-major (o, h)
             const float* __restrict__ bias,   // (128)
             const float* __restrict__ anorm,  // (16, 16)
             const int*   __restrict__ cnts,   // (B, 16)
             const int*   __restrict__ lists,  // (B, 16, 64)
             float*       __restrict__ out) {  // (B, 16, 128)
    __shared__ float Esh[NC][HID];
    __shared__ float Gsh[NC][HID + 4];   // 132 floats/row: bank-conflict-free b64 reads
    __shared__ float Ash[NC * NC];
    __shared__ int   Lsh[NC * CSZ];
    __shared__ int   Csh[NC];

    const int b   = blockIdx.x;
    const int tid = threadIdx.x;

    // Stage metadata into LDS.
    Ash[tid] = anorm[tid];
    {
        const int* lp = lists + (size_t)b * (NC * CSZ);
        #pragma unroll
        for (int i = tid; i < NC * CSZ; i += 256) Lsh[i] = lp[i];
    }
    if (tid < NC) Csh[tid] = cnts[b * NC + tid];
    __syncthreads();

    // ---- Phase B: max-pool core embeddings (2 cores at a time, h = tid&127)
    const int hh    = tid & (HID - 1);
    const int cpart = tid >> 7;          // 0 or 1
    for (int seg = 0; seg < 8; ++seg) {
        const int core = seg * 2 + cpart;
        const int cnt  = Csh[core];
        const int m    = cnt < CSZ ? cnt : CSZ;
        const int* lst = &Lsh[core * CSZ];
        float acc = -INFINITY;
        for (int s = 0; s < m; ++s) {
            const int q = lst[s];                       // LDS broadcast
            acc = fmaxf(acc, emb[q * HID + hh]);        // coalesced row read (L2)
        }
        if (cnt < CSZ) acc = fmaxf(acc, 0.0f);          // zero-padding rows
        Esh[core][hh] = acc;
    }
    __syncthreads();

    // ---- Phase G: G = a_norm @ E  (16x128, 8 elements per thread)
    #pragma unroll
    for (int r = 0; r < 8; ++r) {
        const int idx = tid + r * 256;    // 0..2047
        const int i = idx >> 7, h = idx & (HID - 1);
        float s = 0.f;
        #pragma unroll
        for (int j = 0; j < NC; ++j) s = fmaf(Ash[i * NC + j], Esh[j][h], s);
        Gsh[i][h] = s;
    }
    __syncthreads();

    // ---- Phase C: out_b = G @ W^T + bias, fp32 WMMA 16x16x4, wave-per-N-tile.
    // A (16x4 f32) layout: lane<16 -> M=lane, K=0,1 in v[0],v[1];
    //                      lane>=16 -> M=lane-16, K=2,3.  (ISA 7.12.2)
    // B (4x16 f32) mirrors with N in place of M. B[k][n] = W[N0+n][4k+k'].
    const int lane = tid & 31;
    const int wave = tid >> 5;           // 0..7
    const int N0   = wave * 16;
    const int mrow = lane & 15;
    const int ko   = (lane >> 4) * 2;    // 0 or 2
    const float* wbase = W + (size_t)(N0 + mrow) * HID + ko;

    v8f acc = {};
    #pragma unroll 4
    for (int k = 0; k < HID / 4; ++k) {
        v2f a, bb;
        a.x  = Gsh[mrow][4 * k + ko];
        a.y  = Gsh[mrow][4 * k + ko + 1];
        bb.x = wbase[4 * k];
        bb.y = wbase[4 * k + 1];
        acc = __builtin_amdgcn_wmma_f32_16x16x4_f32(
                  /*neg_a=*/false, a, /*neg_b=*/false, bb,
                  /*c_mod=*/(short)0, acc, /*reuse_a=*/false, /*reuse_b=*/false);
    }

    // C/D layout: VGPR r -> M=r (lanes 0-15) or M=r+8 (lanes 16-31); N = lane&15.
    const int half = lane >> 4;
    const int ncol = N0 + (lane & 15);
    const float bv = bias[ncol];
    float* obase = out + (size_t)b * (NC * HID) + ncol;
    #pragma unroll
    for (int r = 0; r < 8; ++r) {
        const int M = r + half * 8;
        obase[M * HID] = acc[r] + bv;
    }
}

// ---------------------------------------------------------------------------
extern "C" void kernel_launch(void* const* d_in, const int* in_sizes, int n_in,
                              void* d_out, int out_size, void* d_ws, size_t ws_size,
                              hipStream_t stream) {
    (void)n_in; (void)out_size; (void)ws_size;

    const int*   assign = (const int*)  d_in[0];   // (B, 512) int32
    const float* adj    = (const float*)d_in[1];   // (16, 16)
    const float* emb    = (const float*)d_in[2];   // (513, 128)
    const float* W      = (const float*)d_in[3];   // (128, 128)
    const float* bias   = (const float*)d_in[4];   // (128)
    float*       out    = (float*)d_out;           // (B, 16, 128)

    const int B = in_sizes[0] / NQ;                // 2048

    // Workspace layout (~8.5 MB): anorm | cnts | lists
    char* ws = (char*)d_ws;
    float* anorm = (float*)ws;                          // 16*16 f32   (1 KB)
    int*   cnts  = (int*)(ws + 1024);                   // B*16 i32    (128 KB)
    int*   lists = (int*)(ws + 1024 + (size_t)B * NC * sizeof(int)); // B*16*64 i32 (8 MB)

    anorm_kernel  <<<1, 256, 0, stream>>>(adj, anorm);
    scatter_kernel<<<B, 32,  0, stream>>>(assign, cnts, lists);
    fused_kernel  <<<B, 256, 0, stream>>>(emb, W, bias, anorm, cnts, lists, out);
}